// ITSCLLoss_46935402610856
// MI455X (gfx1250) — compile-verified
//
#include <hip/hip_runtime.h>
#include <hip/hip_bf16.h>
#include <math.h>

// ---------------- problem constants (from reference) ----------------
#define Bsz   32
#define Sseq  2048
#define Hdim  1024
#define Ccls  5
#define TEMP  0.07f
#define EPSL  1e-8f

#define SSPLIT 16
#define SCHUNK (Sseq / SSPLIT)   // 128

// ---------------- workspace layout (float offsets) ----------------
#define OFF_WA    ((size_t)0)                       // [B,S]      65536
#define OFF_WO    (OFF_WA + (size_t)Bsz*Sseq)       // [B,S]      65536
#define OFF_PA    (OFF_WO + (size_t)Bsz*Sseq)       // [SSPLIT,B,H] 524288
#define OFF_PO    (OFF_PA + (size_t)SSPLIT*Bsz*Hdim)
#define OFF_SEMB  (OFF_PO + (size_t)SSPLIT*Bsz*Hdim) // [B,2H]    65536
#define OFF_HA    (OFF_SEMB + (size_t)Bsz*2*Hdim)    // [B,512]
#define OFF_HO    (OFF_HA + (size_t)Bsz*512)
#define OFF_HS    (OFF_HO + (size_t)Bsz*512)         // [B,1024]
#define OFF_AP    (OFF_HS + (size_t)Bsz*1024)        // [B,128]
#define OFF_OP    (OFF_AP + (size_t)Bsz*128)
#define OFF_SP    (OFF_OP + (size_t)Bsz*128)         // [B,256]

typedef __attribute__((ext_vector_type(2))) float v2f;
typedef __attribute__((ext_vector_type(8))) float v8f;

// ---------------- 1) softmax weights over S (both logit sets) ----------------
__global__ __launch_bounds__(256) void k_softmax_w(const float* __restrict__ alog,
                                                   const float* __restrict__ olog,
                                                   float* __restrict__ w_a,
                                                   float* __restrict__ w_o) {
    const int b = blockIdx.x;
    __shared__ float tbuf[Sseq];
    __shared__ float red[256];
    for (int pass = 0; pass < 2; ++pass) {
        const float* lg = pass ? olog : alog;
        float* w = pass ? w_o : w_a;
        float lmax = -INFINITY;
        for (int s = threadIdx.x; s < Sseq; s += 256) {
            const float* p = lg + ((size_t)b * Sseq + s) * Ccls;
            float tv = p[1] + p[2] + p[3] + p[4];   // sum of non-O classes
            tbuf[s] = tv;
            lmax = fmaxf(lmax, tv);
        }
        red[threadIdx.x] = lmax;
        __syncthreads();
        for (int off = 128; off > 0; off >>= 1) {
            if (threadIdx.x < off) red[threadIdx.x] = fmaxf(red[threadIdx.x], red[threadIdx.x + off]);
            __syncthreads();
        }
        const float m = red[0];
        __syncthreads();
        float lsum = 0.f;
        for (int s = threadIdx.x; s < Sseq; s += 256) {
            float e = expf(tbuf[s] - m);
            tbuf[s] = e;
            lsum += e;
        }
        red[threadIdx.x] = lsum;
        __syncthreads();
        for (int off = 128; off > 0; off >>= 1) {
            if (threadIdx.x < off) red[threadIdx.x] += red[threadIdx.x + off];
            __syncthreads();
        }
        const float inv = 1.0f / red[0];
        __syncthreads();
        for (int s = threadIdx.x; s < Sseq; s += 256)
            w[(size_t)b * Sseq + s] = tbuf[s] * inv;
        __syncthreads();
    }
}

// ---------------- 2) fused dual attention-pooling (single pass over 256MB) ----
__global__ __launch_bounds__(256) void k_pool_partial(const float* __restrict__ span,
                                                      const float* __restrict__ w_a,
                                                      const float* __restrict__ w_o,
                                                      float* __restrict__ pa,
                                                      float* __restrict__ po) {
    const int b = blockIdx.x, sc = blockIdx.y;
    const int s0 = sc * SCHUNK;
    __shared__ float swa[SCHUNK], swo[SCHUNK];
    if (threadIdx.x < SCHUNK)
        swa[threadIdx.x] = w_a[(size_t)b * Sseq + s0 + threadIdx.x];
    else
        swo[threadIdx.x - SCHUNK] = w_o[(size_t)b * Sseq + s0 + (threadIdx.x - SCHUNK)];
    __syncthreads();
    const int h4 = threadIdx.x * 4;                 // 256 threads * 4 = full H row
    const float* base = span + ((size_t)b * Sseq + s0) * Hdim + h4;
    float4 sa = {0.f, 0.f, 0.f, 0.f}, so = {0.f, 0.f, 0.f, 0.f};
    for (int s = 0; s < SCHUNK; ++s) {
        const float4 v = *(const float4*)(base + (size_t)s * Hdim);  // b128 coalesced
        const float fa = swa[s], fo = swo[s];
        sa.x += v.x * fa; sa.y += v.y * fa; sa.z += v.z * fa; sa.w += v.w * fa;
        so.x += v.x * fo; so.y += v.y * fo; so.z += v.z * fo; so.w += v.w * fo;
    }
    *(float4*)(pa + ((size_t)sc * Bsz + b) * Hdim + h4) = sa;
    *(float4*)(po + ((size_t)sc * Bsz + b) * Hdim + h4) = so;
}

__global__ __launch_bounds__(256) void k_pool_reduce(const float* __restrict__ pa,
                                                     const float* __restrict__ po,
                                                     float* __restrict__ sEmb) {
    const int b = blockIdx.x;
    const int h4 = threadIdx.x * 4;
    float4 sa = {0.f, 0.f, 0.f, 0.f}, so = {0.f, 0.f, 0.f, 0.f};
    for (int c = 0; c < SSPLIT; ++c) {
        const float4 va = *(const float4*)(pa + ((size_t)c * Bsz + b) * Hdim + h4);
        const float4 vo = *(const float4*)(po + ((size_t)c * Bsz + b) * Hdim + h4);
        sa.x += va.x; sa.y += va.y; sa.z += va.z; sa.w += va.w;
        so.x += vo.x; so.y += vo.y; so.z += vo.z; so.w += vo.w;
    }
    *(float4*)(sEmb + (size_t)b * (2 * Hdim) + h4) = sa;          // a_emb
    *(float4*)(sEmb + (size_t)b * (2 * Hdim) + Hdim + h4) = so;   // o_emb
}

// ---------------- 3) fp32 WMMA GEMM: OUT[32,N] = act(X[32,K] @ W[K,N] + bias) --
// One wave per 16-wide N strip; two M-tiles share each B fragment.
// V_WMMA_F32_16X16X4_F32 per ISA 7.12.2 32-bit layouts.
__global__ __launch_bounds__(32) void k_gemm_wmma_f32(const float* __restrict__ X, int ldx,
                                                      const float* __restrict__ W,
                                                      const float* __restrict__ bias,
                                                      float* __restrict__ OUT,
                                                      int K, int N, int relu) {
    const int lane  = threadIdx.x;      // 0..31, full wave, EXEC all-ones
    const int nbase = blockIdx.x * 16;
    const int mlo   = lane & 15;
    const int khalf = lane >> 4;        // 0 -> K{0,1}, 1 -> K{2,3}
    v8f acc0 = {}; v8f acc1 = {};
    for (int k0 = 0; k0 < K; k0 += 4) {
        const int ka = k0 + 2 * khalf;
        v2f bfrag;                      // B 4x16: row ka across lanes within vgpr
        bfrag.x = W[(size_t)ka * N + nbase + mlo];
        bfrag.y = W[(size_t)(ka + 1) * N + nbase + mlo];
        v2f a0, a1;                     // A 16x4: row mlo, K pair per lane-half
        a0.x = X[(size_t)mlo * ldx + ka];
        a0.y = X[(size_t)mlo * ldx + ka + 1];
        a1.x = X[(size_t)(mlo + 16) * ldx + ka];
        a1.y = X[(size_t)(mlo + 16) * ldx + ka + 1];
        acc0 = __builtin_amdgcn_wmma_f32_16x16x4_f32(false, a0, false, bfrag,
                                                     (short)0, acc0, false, false);
        acc1 = __builtin_amdgcn_wmma_f32_16x16x4_f32(false, a1, false, bfrag,
                                                     (short)0, acc1, false, false);
    }
    const float bb = bias[nbase + mlo];
#pragma unroll
    for (int r = 0; r < 8; ++r) {
        float v0 = acc0[r] + bb;
        float v1 = acc1[r] + bb;
        if (relu) { v0 = fmaxf(v0, 0.f); v1 = fmaxf(v1, 0.f); }
        const int row = r + 8 * khalf;  // D layout: row = vgpr + 8*(lane>=16)
        OUT[(size_t)row * N + nbase + mlo] = v0;
        OUT[(size_t)(16 + row) * N + nbase + mlo] = v1;
    }
}

// ---------------- 4) row L2 normalize (in place) ----------------
__global__ __launch_bounds__(128) void k_l2norm(float* __restrict__ V, int n) {
    const int row = blockIdx.x;
    __shared__ float red[128];
    float s = 0.f;
    for (int i = threadIdx.x; i < n; i += 128) {
        const float x = V[(size_t)row * n + i];
        s += x * x;
    }
    red[threadIdx.x] = s;
    __syncthreads();
    for (int off = 64; off > 0; off >>= 1) {
        if (threadIdx.x < off) red[threadIdx.x] += red[threadIdx.x + off];
        __syncthreads();
    }
    const float inv = 1.0f / fmaxf(sqrtf(red[0]), 1e-12f);
    __syncthreads();
    for (int i = threadIdx.x; i < n; i += 128)
        V[(size_t)row * n + i] *= inv;
}

// ---------------- 5) InfoNCE + NT-Xent loss (single block) ----------------
__global__ __launch_bounds__(1024) void k_loss(const float* __restrict__ aP,
                                               const float* __restrict__ oP,
                                               const float* __restrict__ sP,
                                               const int* __restrict__ labels,
                                               float* __restrict__ out) {
    __shared__ float eAO[Bsz * Bsz];
    __shared__ float eSS[Bsz * Bsz];
    __shared__ int   lab[Bsz];
    __shared__ float rowInfo[Bsz], rowNt[Bsz], rowHas[Bsz];
    const int t = threadIdx.x;
    if (t < Bsz) lab[t] = labels[t];
    __syncthreads();
    const int i = t >> 5, j = t & 31;
    float d = 0.f;
    for (int k = 0; k < 128; ++k) d += aP[i * 128 + k] * oP[j * 128 + k];
    eAO[i * Bsz + j] = expf(d / TEMP);
    float d2 = 0.f;
    for (int k = 0; k < 256; ++k) d2 += sP[i * 256 + k] * sP[j * 256 + k];
    eSS[i * Bsz + j] = expf(d2 / TEMP);
    __syncthreads();
    if (t < Bsz) {
        const int li = lab[t];
        // InfoNCE row
        float num = 0.f, den = 0.f;
        for (int jj = 0; jj < Bsz; ++jj) {
            const float e = eAO[t * Bsz + jj];
            den += e;
            const float pm = (jj == t) ? 1.0f : ((lab[jj] == li) ? 0.5f : 0.0f);
            num += e * pm;
        }
        rowInfo[t] = -logf(num / den + EPSL);
        // NT-Xent row (diagonal counted in neg, as in reference)
        float pos = 0.f, neg = 0.f, pc = 0.f;
        for (int jj = 0; jj < Bsz; ++jj) {
            const float e = eSS[t * Bsz + jj];
            const float m = ((lab[jj] == li) && (jj != t)) ? 1.0f : 0.0f;
            pos += e * m;
            neg += e * (1.0f - m);
            pc  += m;
        }
        const bool has = pc > 0.f;
        rowNt[t]  = has ? (-logf(pos / (pos + neg) + EPSL) / fmaxf(pc, 1.0f)) : 0.0f;
        rowHas[t] = has ? 1.0f : 0.0f;
    }
    __syncthreads();
    if (t == 0) {
        float inf = 0.f, nt = 0.f, h = 0.f;
        for (int r = 0; r < Bsz; ++r) { inf += rowInfo[r]; nt += rowNt[r]; h += rowHas[r]; }
        inf *= (1.0f / Bsz);
        nt = nt / fmaxf(h, 1.0f);
        out[0] = 1.0f * inf + 0.5f * nt;   // LAMBDA_INFONCE, LAMBDA_NTXENT
    }
}

// ---------------- launch ----------------
extern "C" void kernel_launch(void* const* d_in, const int* in_sizes, int n_in,
                              void* d_out, int out_size, void* d_ws, size_t ws_size,
                              hipStream_t stream) {
    const float* span   = (const float*)d_in[0];
    const float* alog   = (const float*)d_in[1];
    const float* olog   = (const float*)d_in[2];
    const int*   labels = (const int*)d_in[3];
    const float* Wa1 = (const float*)d_in[4];
    const float* ba1 = (const float*)d_in[5];
    const float* Wa2 = (const float*)d_in[6];
    const float* ba2 = (const float*)d_in[7];
    const float* Wo1 = (const float*)d_in[8];
    const float* bo1 = (const float*)d_in[9];
    const float* Wo2 = (const float*)d_in[10];
    const float* bo2 = (const float*)d_in[11];
    const float* Ws1 = (const float*)d_in[12];
    const float* bs1 = (const float*)d_in[13];
    const float* Ws2 = (const float*)d_in[14];
    const float* bs2 = (const float*)d_in[15];

    float* ws    = (float*)d_ws;
    float* w_a   = ws + OFF_WA;
    float* w_o   = ws + OFF_WO;
    float* pa    = ws + OFF_PA;
    float* po    = ws + OFF_PO;
    float* sEmb  = ws + OFF_SEMB;   // [B, 2H]: a_emb | o_emb (concat for Ws1)
    float* hA    = ws + OFF_HA;
    float* hO    = ws + OFF_HO;
    float* hS    = ws + OFF_HS;
    float* aProj = ws + OFF_AP;
    float* oProj = ws + OFF_OP;
    float* sProj = ws + OFF_SP;
    float* outF  = (float*)d_out;

    // 1) softmax weights for both aspect/opinion
    k_softmax_w<<<Bsz, 256, 0, stream>>>(alog, olog, w_a, w_o);
    // 2) dual pooling: one pass over span_features (the 256MB bandwidth term)
    k_pool_partial<<<dim3(Bsz, SSPLIT), 256, 0, stream>>>(span, w_a, w_o, pa, po);
    k_pool_reduce<<<Bsz, 256, 0, stream>>>(pa, po, sEmb);
    // 3) MLPs via fp32 WMMA
    k_gemm_wmma_f32<<<512 / 16, 32, 0, stream>>>(sEmb,            2 * Hdim, Wa1, ba1, hA, Hdim,     512,  1);
    k_gemm_wmma_f32<<<512 / 16, 32, 0, stream>>>(sEmb + Hdim,     2 * Hdim, Wo1, bo1, hO, Hdim,     512,  1);
    k_gemm_wmma_f32<<<1024 / 16, 32, 0, stream>>>(sEmb,           2 * Hdim, Ws1, bs1, hS, 2 * Hdim, 1024, 1);
    k_gemm_wmma_f32<<<128 / 16, 32, 0, stream>>>(hA, 512,  Wa2, ba2, aProj, 512,  128, 0);
    k_gemm_wmma_f32<<<128 / 16, 32, 0, stream>>>(hO, 512,  Wo2, bo2, oProj, 512,  128, 0);
    k_gemm_wmma_f32<<<256 / 16, 32, 0, stream>>>(hS, 1024, Ws2, bs2, sProj, 1024, 256, 0);
    // 4) L2 normalize projections
    k_l2norm<<<Bsz, 128, 0, stream>>>(aProj, 128);
    k_l2norm<<<Bsz, 128, 0, stream>>>(oProj, 128);
    k_l2norm<<<Bsz, 128, 0, stream>>>(sProj, 256);
    // 5) losses -> scalar
    k_loss<<<1, 1024, 0, stream>>>(aProj, oProj, sProj, labels, outF);
}